// GP_Neuron_66013647340159
// MI455X (gfx1250) — compile-verified
//
#include <hip/hip_runtime.h>
#include <math.h>

// CDNA5 wave32 WMMA operand types: 16x16x4 f32 -> A/B are 2 VGPRs (float2), C/D 8 VGPRs.
typedef __attribute__((ext_vector_type(2))) float v2f;
typedef __attribute__((ext_vector_type(8))) float v8f;

#define NP 5
#define GP_JITTER 1e-6f
#define TWO_PI_F 6.283185307179586f

__global__ __launch_bounds__(256) void gp_neuron_wmma_kernel(
    const float* __restrict__ x_mean, const float* __restrict__ x_var,
    const float* __restrict__ h_in, const float* __restrict__ z_in,
    const float* __restrict__ l_in, const float* __restrict__ s_in,
    const float* __restrict__ a_in, const float* __restrict__ b_in,
    float* __restrict__ out_mean, float* __restrict__ out_var, int n)
{
    // LDS-resident constants: M1[16][8] WMMA A-matrix + scalars.
    __shared__ float sM1[16][8];
    __shared__ float sZ[8];
    __shared__ float sC[8];   // l2, a, b, c1 = s^2*l, c2 = sqrt(2pi)*s^2*l

    if (threadIdx.x == 0) {
        const float l0 = l_in[0], s0 = s_in[0];
        const float l2 = l0 * l0;
        float zv[NP], hv[NP];
        for (int i = 0; i < NP; ++i) { zv[i] = z_in[i]; hv[i] = h_in[i]; }

        // Q_hh[i][j] = N(z_i; z_j, l^2) + jitter*I
        float Q[NP][NP];
        const float norm = rsqrtf(TWO_PI_F * l2);
        for (int i = 0; i < NP; ++i)
            for (int j = 0; j < NP; ++j) {
                float d = zv[i] - zv[j];
                Q[i][j] = __expf(-(d * d) / (2.0f * l2)) * norm;
            }
        for (int i = 0; i < NP; ++i) Q[i][i] += GP_JITTER;

        // Cholesky: Q = Lm Lm^T (lower)
        float Lm[NP][NP] = {};
        for (int j = 0; j < NP; ++j) {
            float sum = Q[j][j];
            for (int k = 0; k < j; ++k) sum -= Lm[j][k] * Lm[j][k];
            Lm[j][j] = sqrtf(sum);
            float inv = 1.0f / Lm[j][j];
            for (int i = j + 1; i < NP; ++i) {
                float t = Q[i][j];
                for (int k = 0; k < j; ++k) t -= Lm[i][k] * Lm[j][k];
                Lm[i][j] = t * inv;
            }
        }
        // Lower-triangular inverse Li = Lm^-1
        float Li[NP][NP] = {};
        for (int j = 0; j < NP; ++j) {
            Li[j][j] = 1.0f / Lm[j][j];
            for (int i = j + 1; i < NP; ++i) {
                float t = 0.0f;
                for (int k = j; k < i; ++k) t += Lm[i][k] * Li[k][j];
                Li[i][j] = -t / Lm[i][i];
            }
        }
        // w = Li^T (Li h)
        float tv[NP], w[NP];
        for (int i = 0; i < NP; ++i) {
            float t = 0.0f;
            for (int k = 0; k <= i; ++k) t += Li[i][k] * hv[k];
            tv[i] = t;
        }
        for (int k = 0; k < NP; ++k) {
            float t = 0.0f;
            for (int i = k; i < NP; ++i) t += Li[i][k] * tv[i];
            w[k] = t;
        }
        // Pack M1: rows 0..4 = Li rows, row 5 = w, rest zero (K cols 5..7 zero pad).
        for (int m = 0; m < 16; ++m)
            for (int k = 0; k < 8; ++k) sM1[m][k] = 0.0f;
        for (int m = 0; m < NP; ++m)
            for (int k = 0; k < NP; ++k) sM1[m][k] = Li[m][k];
        for (int k = 0; k < NP; ++k) sM1[5][k] = w[k];

        for (int k = 0; k < 8; ++k) sZ[k] = (k < NP) ? zv[k] : 0.0f;
        sC[0] = l2;
        sC[1] = a_in[0];
        sC[2] = b_in[0];
        sC[3] = s0 * s0 * l0;                 // c1
        sC[4] = sqrtf(TWO_PI_F) * s0 * s0 * l0; // c2
    }
    __syncthreads();

    const int lane = threadIdx.x & 31;
    const int hf   = lane >> 4;      // half-wave index (K-group select)
    const int nn   = lane & 15;      // element-in-tile / A-matrix row
    const int waveInBlock = threadIdx.x >> 5;
    const long long base = ((long long)blockIdx.x * (blockDim.x >> 5) + waveInBlock) * 32LL;

    // Constant A-matrix operand regs (16x4 f32 layout: lane L holds row L%16,
    // K = {2*(L/16), 2*(L/16)+1}); A2 covers global K = 4..7.
    const int k0 = 2 * hf;
    v2f A1, A2;
    A1[0] = sM1[nn][k0];     A1[1] = sM1[nn][k0 + 1];
    A2[0] = sM1[nn][4 + k0]; A2[1] = sM1[nn][5 + k0];

    const float l2 = sC[0], aa = sC[1], bb = sC[2], c1 = sC[3], c2 = sC[4];
    const float z0 = sZ[0], z1 = sZ[1], z2 = sZ[2], z3 = sZ[3], z4 = sZ[4];
    const float RSQRT_2PI = 0.3989422804014327f;

    const long long idxA = base + nn;        // tile 0, element nn
    const long long idxB = idxA + 16;        // tile 1, element nn

    float xm0 = 0.0f, xv0 = 1.0f, xm1 = 0.0f, xv1 = 1.0f;
    if (idxA < n) { xm0 = x_mean[idxA]; xv0 = x_var[idxA]; }
    if (idxB < n) { xm1 = x_mean[idxB]; xv1 = x_var[idxB]; }

    // B operand (4x16 f32, K striped across half-waves like A):
    // lanes 0-15 supply q cols {0,1} (B1) and {4, pad} (B2);
    // lanes 16-31 supply q cols {2,3} (B1) and {pad, pad} (B2).
    const float za = hf ? z2 : z0;
    const float zb = hf ? z3 : z1;

    v2f B1t0, B2t0, B1t1, B2t1;
    {
        float var1 = xv0 + l2;
        float r    = rsqrtf(var1);
        float pref = RSQRT_2PI * r;          // 1/sqrt(2*pi*var1)
        float inv2 = -0.5f * r * r;          // -1/(2*var1)
        float d0 = xm0 - za, d1 = xm0 - zb, d4 = xm0 - z4;
        B1t0[0] = pref * __expf(inv2 * d0 * d0);
        B1t0[1] = pref * __expf(inv2 * d1 * d1);
        float q4 = pref * __expf(inv2 * d4 * d4);
        B2t0[0] = hf ? 0.0f : q4;
        B2t0[1] = 0.0f;
    }
    {
        float var1 = xv1 + l2;
        float r    = rsqrtf(var1);
        float pref = RSQRT_2PI * r;
        float inv2 = -0.5f * r * r;
        float d0 = xm1 - za, d1 = xm1 - zb, d4 = xm1 - z4;
        B1t1[0] = pref * __expf(inv2 * d0 * d0);
        B1t1[1] = pref * __expf(inv2 * d1 * d1);
        float q4 = pref * __expf(inv2 * d4 * d4);
        B2t1[0] = hf ? 0.0f : q4;
        B2t1[1] = 0.0f;
    }

    // D = M1 x q^T : column n of D (lane-local in lanes 0..15) gives
    // D[0..4] = A_ref[n,:] (= q @ Linv^T) and D[5] = q . w.
    // EXEC is all-ones here (no early returns, guards are load/store-local).
    v8f acc0 = {};
    acc0 = __builtin_amdgcn_wmma_f32_16x16x4_f32(false, A1, false, B1t0,
                                                 (short)0, acc0, false, false);
    acc0 = __builtin_amdgcn_wmma_f32_16x16x4_f32(false, A2, false, B2t0,
                                                 (short)0, acc0, false, false);
    v8f acc1 = {};
    acc1 = __builtin_amdgcn_wmma_f32_16x16x4_f32(false, A1, false, B1t1,
                                                 (short)0, acc1, false, false);
    acc1 = __builtin_amdgcn_wmma_f32_16x16x4_f32(false, A2, false, B2t1,
                                                 (short)0, acc1, false, false);

    // Lane-local epilogue (valid in lanes 0..15; lanes 16..31 hold zero rows).
    float ss0 = acc0[0]*acc0[0] + acc0[1]*acc0[1] + acc0[2]*acc0[2]
              + acc0[3]*acc0[3] + acc0[4]*acc0[4];
    float qw0 = acc0[5];
    float ss1 = acc1[0]*acc1[0] + acc1[1]*acc1[1] + acc1[2]*acc1[2]
              + acc1[3]*acc1[3] + acc1[4]*acc1[4];
    float qw1 = acc1[5];

    float mean0 = fmaf(aa, xm0, bb) + qw0;
    float varr0 = c1 * rsqrtf(l2 + 2.0f * xv0) - c2 * ss0 + GP_JITTER;
    float mean1 = fmaf(aa, xm1, bb) + qw1;
    float varr1 = c1 * rsqrtf(l2 + 2.0f * xv1) - c2 * ss1 + GP_JITTER;

    // Move tile-1 results into the upper half-wave for one fully coalesced store.
    float mean1b = __shfl(mean1, nn, 32);
    float varr1b = __shfl(varr1, nn, 32);
    float om = (lane < 16) ? mean0 : mean1b;
    float ov = (lane < 16) ? varr0 : varr1b;

    const long long sidx = base + lane;
    if (sidx < n) {
        out_mean[sidx] = om;
        out_var[sidx]  = ov;
    }
}

extern "C" void kernel_launch(void* const* d_in, const int* in_sizes, int n_in,
                              void* d_out, int out_size, void* d_ws, size_t ws_size,
                              hipStream_t stream) {
    const float* x_mean = (const float*)d_in[0];
    const float* x_var  = (const float*)d_in[1];
    const float* h      = (const float*)d_in[2];
    const float* z      = (const float*)d_in[3];
    const float* l      = (const float*)d_in[4];
    const float* s      = (const float*)d_in[5];
    const float* a      = (const float*)d_in[6];
    const float* b      = (const float*)d_in[7];

    const int n = in_sizes[0];
    float* out_mean = (float*)d_out;
    float* out_var  = out_mean + n;

    const int threads = 256;                       // 8 wave32 waves, 1 element/lane
    const int blocks  = (n + threads - 1) / threads;
    gp_neuron_wmma_kernel<<<blocks, threads, 0, stream>>>(
        x_mean, x_var, h, z, l, s, a, b, out_mean, out_var, n);
}